// Gconv_55413668053696
// MI455X (gfx1250) — compile-verified
//
#include <hip/hip_runtime.h>
#include <hip/hip_bf16.h>

typedef float v2f __attribute__((ext_vector_type(2)));
typedef float v8f __attribute__((ext_vector_type(8)));

#define BS 8
#define NN 1024
#define JJ 3
#define FF 64
#define CC 64
#define NIN 256   // (J+1)*F

// LDS row paddings chosen so the per-lane bank stride is 4 words
// (68 % 64 == 4, 260 % 64 == 4) -> conflict-free b64 fragment loads.
#define APAD 68
#define XPAD 68
#define HPAD 260

// ---------------- Kernel 1: gmul (WMMA f32) + fused W copy + FC (WMMA f32) ----
// grid: (64, 8) = (m_tile, b), block: 256 threads (8 waves)
struct StageLDS {
    float A[4][16][APAD];   // W_all staged as [k][m][n]   (~17.4 KB)
    float XT[64][XPAD];     // x staged transposed [f][n]  (~17.4 KB)
};
union GconvLDS {
    StageLDS s;
    float H[16][HPAD];      // h tile [m][k*64+f]          (~16.6 KB)
};

__global__ __launch_bounds__(256) void gconv_kernel(
    const float* __restrict__ W, const float* __restrict__ x,
    const float* __restrict__ y,
    const float* __restrict__ fc1_w, const float* __restrict__ fc1_b,
    const float* __restrict__ fc2_w, const float* __restrict__ fc2_b,
    float* __restrict__ outW, float* __restrict__ xo_pre)
{
    __shared__ GconvLDS lds;

    const int m_tile = blockIdx.x;
    const int b      = blockIdx.y;
    const int m_base = m_tile * 16;
    const int t      = threadIdx.x;
    const int wave   = t >> 5;
    const int lane   = t & 31;

    const int k      = wave >> 1;        // 0..3 (k==3 is the y slice)
    const int f_base = (wave & 1) * 32;  // 0 or 32

    // per-lane WMMA addressing
    const int mA  = lane & 15;                 // A row / H row
    const int kkA = (lane < 16) ? 0 : 2;       // K pair selector for half-wave
    const int nB  = lane & 15;                 // B/D column within tile
    const int mD  = (lane >= 16) ? 8 : 0;      // D row offset

    v8f acc0 = {}; v8f acc1 = {};

    for (int nc = 0; nc < 16; ++nc) {
        const int n_base = nc * 64;

        // ---- stage W_all tile: thread t handles (m = t>>4, n = (t&15)*4 .. +3)
        {
            const int m  = t >> 4;
            const int nn = (t & 15) * 4;
            const size_t row = ((size_t)b * NN + m_base + m) * NN + n_base + nn;
            const float4* wp = reinterpret_cast<const float4*>(W + row * 3); // 12 floats, 16B aligned
            float4 w0 = wp[0], w1 = wp[1], w2 = wp[2];
            float4 yv = *reinterpret_cast<const float4*>(y + row);

            // fused passthrough copy of W into d_out
            float4* op = reinterpret_cast<float4*>(outW + row * 3);
            op[0] = w0; op[1] = w1; op[2] = w2;

            float wf[12] = {w0.x, w0.y, w0.z, w0.w, w1.x, w1.y, w1.z, w1.w,
                            w2.x, w2.y, w2.z, w2.w};
            float yf[4]  = {yv.x, yv.y, yv.z, yv.w};
#pragma unroll
            for (int i = 0; i < 4; ++i) {
                lds.s.A[0][m][nn + i] = wf[i * 3 + 0];
                lds.s.A[1][m][nn + i] = wf[i * 3 + 1];
                lds.s.A[2][m][nn + i] = wf[i * 3 + 2];
                lds.s.A[3][m][nn + i] = yf[i];
            }
        }
        // ---- stage x tile transposed: thread t handles (n = t>>2, f = (t&3)*16 .. +15)
        {
            const int n  = t >> 2;
            const int f0 = (t & 3) * 16;
            const float4* xp = reinterpret_cast<const float4*>(
                x + ((size_t)b * NN + n_base + n) * FF + f0);
#pragma unroll
            for (int i = 0; i < 4; ++i) {
                float4 v = xp[i];
                lds.s.XT[f0 + i * 4 + 0][n] = v.x;
                lds.s.XT[f0 + i * 4 + 1][n] = v.y;
                lds.s.XT[f0 + i * 4 + 2][n] = v.z;
                lds.s.XT[f0 + i * 4 + 3][n] = v.w;
            }
        }
        __syncthreads();

        // ---- 16 WMMA K=4 steps over this 64-wide n chunk
#pragma unroll
        for (int nn = 0; nn < 16; ++nn) {
            const int n4 = nn * 4;
            v2f Af = *reinterpret_cast<const v2f*>(&lds.s.A[k][mA][n4 + kkA]);
            v2f B0 = *reinterpret_cast<const v2f*>(&lds.s.XT[f_base + nB][n4 + kkA]);
            v2f B1 = *reinterpret_cast<const v2f*>(&lds.s.XT[f_base + 16 + nB][n4 + kkA]);
            acc0 = __builtin_amdgcn_wmma_f32_16x16x4_f32(
                false, Af, false, B0, (short)0, acc0, false, false);
            acc1 = __builtin_amdgcn_wmma_f32_16x16x4_f32(
                false, Af, false, B1, (short)0, acc1, false, false);
        }
        __syncthreads();
    }

    // ---- spill h tile [16][256] to LDS (D layout: VGPR r -> row r+mD, col nB)
    {
        const int colbase = k * 64 + f_base;
#pragma unroll
        for (int r = 0; r < 8; ++r) {
            lds.H[r + mD][colbase + nB]      = acc0[r];
            lds.H[r + mD][colbase + 16 + nB] = acc1[r];
        }
    }
    __syncthreads();

    // ---- FC: out[16][64] = H[16][256] @ [fc1_w | fc2_w], waves 0..3 each own a 16-col tile
    if (wave < 4) {
        const int c_base = wave * 16;
        const bool half1 = (c_base < 32);
        const float* Wfc = half1 ? fc1_w : fc2_w;     // [256][32] row-major
        const int cc = (c_base & 31) + nB;            // column within 32-wide weight

        v8f accF = {};
#pragma unroll 4
        for (int r4 = 0; r4 < NIN; r4 += 4) {
            v2f Af = *reinterpret_cast<const v2f*>(&lds.H[mA][r4 + kkA]);
            v2f Bf;
            Bf.x = Wfc[(size_t)(r4 + kkA) * 32 + cc];
            Bf.y = Wfc[(size_t)(r4 + kkA + 1) * 32 + cc];
            accF = __builtin_amdgcn_wmma_f32_16x16x4_f32(
                false, Af, false, Bf, (short)0, accF, false, false);
        }

        const float bias = half1 ? fc1_b[cc] : fc2_b[cc];
        const int c = c_base + nB;
#pragma unroll
        for (int r = 0; r < 8; ++r) {
            float v = accF[r] + bias;
            if (half1) v = fmaxf(v, 0.0f);
            const int m = m_base + r + mD;
            xo_pre[((size_t)b * NN + m) * CC + c] = v;
        }
    }
}

// ---------------- Kernel 2: per (b,c) mean / rstd over N ---------------------
__global__ __launch_bounds__(256) void stats_kernel(
    const float* __restrict__ xo_pre, float* __restrict__ mean,
    float* __restrict__ rstd)
{
    const int b = blockIdx.x;
    const int t = threadIdx.x;
    const int c = t & 63;
    const int seg = t >> 6;    // 0..3, each sums 256 rows

    float s = 0.f, s2 = 0.f;
    for (int n = seg * 256; n < seg * 256 + 256; ++n) {
        float v = xo_pre[((size_t)b * NN + n) * CC + c];
        s += v; s2 += v * v;
    }
    __shared__ float ss[4][64];
    __shared__ float sq[4][64];
    ss[seg][c] = s; sq[seg][c] = s2;
    __syncthreads();
    if (t < 64) {
        float S  = ss[0][c] + ss[1][c] + ss[2][c] + ss[3][c];
        float S2 = sq[0][c] + sq[1][c] + sq[2][c] + sq[3][c];
        float m  = S * (1.0f / NN);
        float var = S2 * (1.0f / NN) - m * m;
        mean[b * 64 + c] = m;
        rstd[b * 64 + c] = rsqrtf(var + 1e-5f);
    }
}

// ---------------- Kernel 3: normalize, write xo, compute bx ------------------
__global__ __launch_bounds__(256) void norm_bx_kernel(
    const float* __restrict__ xo_pre, const float* __restrict__ mean,
    const float* __restrict__ rstd, const float* __restrict__ beta_w,
    const float* __restrict__ beta_b, float* __restrict__ xo_out,
    float* __restrict__ bx)
{
    const int r = blockIdx.x * blockDim.x + threadIdx.x;  // 0..8191 row id
    const int b = r >> 10;
    const float* row = xo_pre + (size_t)r * CC;
    float* orow = xo_out + (size_t)r * CC;
    float acc = beta_b[0];
#pragma unroll 8
    for (int c = 0; c < CC; ++c) {
        float v = (row[c] - mean[b * 64 + c]) * rstd[b * 64 + c];
        orow[c] = v;
        acc += v * beta_w[c];
    }
    bx[r] = acc;
}

// ---------------- Kernel 4: edge update ------------------------------------
__global__ __launch_bounds__(256) void edge_kernel(
    const float* __restrict__ y, const float* __restrict__ bx,
    const float* __restrict__ sigma_p, float* __restrict__ yo)
{
    const size_t idx = ((size_t)blockIdx.x * blockDim.x + threadIdx.x) * 4;
    const int b   = (int)(idx >> 20);
    const int rem = (int)(idx & ((1u << 20) - 1));
    const int m   = rem >> 10;
    const int n0  = rem & 1023;

    const float sg = sigma_p[0];
    const float bm = bx[(b << 10) + m];
    float4 yv = *reinterpret_cast<const float4*>(y + idx);
    float vin[4] = {yv.x, yv.y, yv.z, yv.w};
    float vout[4];
#pragma unroll
    for (int i = 0; i < 4; ++i) {
        float v = bm + bx[(b << 10) + n0 + i] + sg * vin[i];
        float s = 1.0f / (1.0f + __expf(-v));
        vout[i] = (m == n0 + i) ? 0.0f : s;
    }
    float4 o = {vout[0], vout[1], vout[2], vout[3]};
    *reinterpret_cast<float4*>(yo + idx) = o;
}

// ---------------- launch -----------------------------------------------------
extern "C" void kernel_launch(void* const* d_in, const int* in_sizes, int n_in,
                              void* d_out, int out_size, void* d_ws, size_t ws_size,
                              hipStream_t stream)
{
    const float* W      = (const float*)d_in[0];
    const float* x      = (const float*)d_in[1];
    const float* y      = (const float*)d_in[2];
    const float* fc1_w  = (const float*)d_in[3];
    const float* fc1_b  = (const float*)d_in[4];
    const float* fc2_w  = (const float*)d_in[5];
    const float* fc2_b  = (const float*)d_in[6];
    const float* beta_w = (const float*)d_in[7];
    const float* beta_b = (const float*)d_in[8];
    const float* sigma  = (const float*)d_in[9];

    // d_out layout: W copy | xo | yo
    float* outW = (float*)d_out;
    float* outXo = outW + (size_t)BS * NN * NN * JJ;          // 25165824
    float* outYo = outXo + (size_t)BS * NN * CC;              // +524288

    // workspace layout
    char* ws = (char*)d_ws;
    float* xo_pre = (float*)ws;                               // 2 MB
    float* mean   = (float*)(ws + (size_t)BS * NN * CC * 4);
    float* rstd   = mean + BS * 64;
    float* bx     = rstd + BS * 64;                           // 32 KB

    dim3 g1(NN / 16, BS);
    gconv_kernel<<<g1, 256, 0, stream>>>(W, x, y, fc1_w, fc1_b, fc2_w, fc2_b,
                                         outW, xo_pre);
    stats_kernel<<<BS, 256, 0, stream>>>(xo_pre, mean, rstd);
    norm_bx_kernel<<<(BS * NN) / 256, 256, 0, stream>>>(xo_pre, mean, rstd,
                                                        beta_w, beta_b, outXo, bx);
    edge_kernel<<<(BS * NN * NN) / 1024, 256, 0, stream>>>(y, bx, sigma, outYo);
}